// TemporalBlock_15032385536168
// MI455X (gfx1250) — compile-verified
//
#include <hip/hip_runtime.h>
#include <hip/hip_bf16.h>

// ---------------------------------------------------------------------------
// Problem constants (from reference): B=4, T=16, N=1000, Fin=64, Fout=128,
// E=12000, H=4, Dh=32, L=B*T=64, TB=T*B=64.
//
// With H0 = 0 the GConvGRU collapses: conv(h0,*)=bias, R is dead, so
//   Hn = (1 - sigmoid(Xcat@Wz + bz)) * tanh(Xcat@Wh + bh),
//   Xcat = [xg | agg],  agg = sym-normalized 1-hop aggregation.
// MHA then runs per node (batch=N): QKV, 4-head softmax attention, out-proj.
// Every matmul goes through V_WMMA_F32_16X16X4_F32 (fp32 matrix pipe).
// ---------------------------------------------------------------------------
#define NNODE 1000
#define NEDGE 12000
#define TB    64      // merged time-batch dim (t*B+b)
#define LQ    64      // MHA sequence length (b*T+t)
#define FIN   64
#define FOUT  128

typedef float v2f __attribute__((ext_vector_type(2)));
typedef float v8f __attribute__((ext_vector_type(8)));

// V_WMMA_F32_16X16X4_F32:  D(16x16,f32) = A(16x4,f32) * B(4x16,f32) + C
// Lane layout (ISA 7.12.2): A/B element (lane%16, K = vgpr + 2*(lane/16));
// D vgpr i holds (M = i + 8*(lane/16), N = lane%16).
__device__ __forceinline__ v8f wmma4(v2f a, v2f b, v8f c) {
  return __builtin_amdgcn_wmma_f32_16x16x4_f32(false, a, false, b, (short)0, c,
                                               false, false);
}

__device__ __forceinline__ void atomic_addf(float* p, float v) {
  __hip_atomic_fetch_add(p, v, __ATOMIC_RELAXED, __HIP_MEMORY_SCOPE_AGENT);
}

// --------------------------- kernel 1: zero ws ------------------------------
__global__ void k_zero(float* __restrict__ p, int n) {
  int i = blockIdx.x * 256 + threadIdx.x;
  if (i < n) p[i] = 0.0f;
}

// --------------------------- kernel 2: degree -------------------------------
__global__ void k_deg(const int* __restrict__ ei, float* __restrict__ deg) {
  int e = blockIdx.x * 256 + threadIdx.x;
  if (e < NEDGE) atomic_addf(&deg[ei[e]], 1.0f);   // row = ei[0][e]
}

// --------------------------- kernel 3: dinv ---------------------------------
__global__ void k_dinv(float* __restrict__ deg) {
  int n = blockIdx.x * 256 + threadIdx.x;
  if (n < NNODE) {
    float d = deg[n];
    deg[n] = (d > 0.0f) ? rsqrtf(fmaxf(d, 1.0f)) : 0.0f;
  }
}

// ------------------- kernel 4: edge scatter into agg ------------------------
// agg[(row*64 + l')*64 + k] += dinv[row]*dinv[col] * xg[col, l', k]
// xg[n, l', k] = x[((b*16 + t)*1000 + n)*64 + k],  t = l'/4, b = l'%4
__global__ __launch_bounds__(256) void k_scatter(
    const int* __restrict__ ei, const float* __restrict__ x,
    const float* __restrict__ dinv, float* __restrict__ agg) {
  int e = blockIdx.x;
  int r = ei[e];
  int c = ei[NEDGE + e];
  float w = dinv[r] * dinv[c];
  for (int idx = threadIdx.x; idx < TB * FIN; idx += 256) {
    int lp = idx >> 6, k = idx & 63;
    int t = lp >> 2, b = lp & 3;
    float val = x[((size_t)((b * 16 + t) * NNODE + c)) * FIN + k];
    atomic_addf(&agg[(size_t)r * (TB * FIN) + idx], w * val);
  }
}

// ------------------- kernel 5: gate GEMM + GRU epilogue ---------------------
// Rows r = n*64 + l' (M = 64000).  K = 128 = [xg | agg].  Each wave owns one
// 16-col tile and two WMMA accumulators (z gate and h gate).
__global__ __launch_bounds__(256) void k_gates(
    const float* __restrict__ x, const float* __restrict__ agg,
    const float* __restrict__ wx0, const float* __restrict__ wx1,
    const float* __restrict__ bx, const float* __restrict__ bhb,
    float* __restrict__ Hn) {
  const int lane = threadIdx.x & 31;
  const int wv   = threadIdx.x >> 5;          // 8 waves -> 8 col tiles = 128
  const int g    = lane >> 4;                 // half-wave selector (K group)
  const int ln16 = lane & 15;

  const int rowA = blockIdx.x * 16 + ln16;    // A-matrix row this lane feeds
  const int nA = rowA / TB, lpA = rowA % TB;
  const int tA = lpA >> 2, bA = lpA & 3;
  const float* __restrict__ xrow = x + ((size_t)((bA * 16 + tA) * NNODE + nA)) * FIN;
  const float* __restrict__ arow = agg + (size_t)rowA * FIN;

  const int f = wv * 16 + ln16;               // output column (B-matrix col)
  const float* __restrict__ wz0 = wx0 + f;                   // gate 0 self
  const float* __restrict__ wz1 = wx1 + f;                   // gate 0 neighbor
  const float* __restrict__ wh0 = wx0 + 2 * FIN * FOUT + f;  // gate 2 self
  const float* __restrict__ wh1 = wx1 + 2 * FIN * FOUT + f;  // gate 2 neighbor

  v8f accZ = {};
  v8f accH = {};
  for (int ks = 0; ks < 32; ++ks) {
    int k0 = ks * 4 + 2 * g;
    v2f a, bz, bh2;
    if (k0 < FIN) {                            // self (xg) half of K
      a.x = xrow[k0];         a.y = xrow[k0 + 1];
      bz.x = wz0[k0 * FOUT];  bz.y = wz0[(k0 + 1) * FOUT];
      bh2.x = wh0[k0 * FOUT]; bh2.y = wh0[(k0 + 1) * FOUT];
    } else {                                   // aggregated half of K
      int kk = k0 - FIN;
      a.x = arow[kk];         a.y = arow[kk + 1];
      bz.x = wz1[kk * FOUT];  bz.y = wz1[(kk + 1) * FOUT];
      bh2.x = wh1[kk * FOUT]; bh2.y = wh1[(kk + 1) * FOUT];
    }
    accZ = wmma4(a, bz, accZ);
    accH = wmma4(a, bh2, accH);
  }

  const float zb = bx[f] + bhb[f];
  const float hb = bx[2 * FOUT + f] + bhb[2 * FOUT + f];
#pragma unroll
  for (int i = 0; i < 8; ++i) {
    int ml = i + 8 * g;
    int r = blockIdx.x * 16 + ml;
    int n = r / TB, lp = r % TB;
    int t = lp >> 2, b = lp & 3;
    int l = b * 16 + t;                       // MHA sequence index
    float z  = 1.0f / (1.0f + __expf(-(accZ[i] + zb)));
    float ht = tanhf(accH[i] + hb);
    Hn[((size_t)(l * NNODE + n)) * FOUT + f] = (1.0f - z) * ht;
  }
}

// ------------------- kernel 6: fused MHA, all-WMMA --------------------------
// One block per node n.  256 threads = 8 waves.  All four matmuls (QKV,
// scores, ctx, out-proj) run on the fp32 WMMA pipe from LDS operands.
__global__ __launch_bounds__(256) void k_mha(
    const float* __restrict__ Hn, const float* __restrict__ w_in,
    const float* __restrict__ b_in, const float* __restrict__ w_out,
    const float* __restrict__ b_out, float* __restrict__ out) {
  __shared__ float sA[LQ * FOUT];        // staged a rows; reused as ctx
  __shared__ float sQ[LQ * FOUT];        // scaled q:  sQ[l][h*32+d]
  __shared__ float sK[LQ * FOUT];        // k:         sK[l][h*32+d]
  __shared__ float sV[LQ * FOUT];        // v:         sV[l][h*32+d]
  __shared__ float sS[4 * LQ * LQ];      // scores/probs: sS[(h*64+l)*64+m]

  const int n = blockIdx.x;
  const int tid = threadIdx.x;
  const int lane = tid & 31, wv = tid >> 5;
  const int g = lane >> 4, ln16 = lane & 15;
  const float scale = 0.17677669529663687f;  // 1/sqrt(Dh=32)

  // phase 1: stage a[l, n, :] (l = 0..63)
  for (int i = tid; i < LQ * FOUT; i += 256) {
    int l = i >> 7, fc = i & 127;
    sA[i] = Hn[((size_t)(l * NNODE + n)) * FOUT + fc];
  }
  __syncthreads();

  // phase 2: QKV = a @ w_in^T + b_in   (96 16x16 tiles, 12 per wave)
  // B[k][j] = w_in[j*128 + k]
  for (int r = 0; r < 12; ++r) {
    int tt = wv + 8 * r;                 // tile id 0..95
    int mt = tt & 3, ct = tt >> 2;       // ct 0..23 -> j segment
    int col = ct * 16 + ln16;            // qkv column j
    const float* __restrict__ arow = sA + (mt * 16 + ln16) * FOUT;
    const float* __restrict__ brow = w_in + (size_t)col * FOUT;
    v8f acc = {};
    for (int ks = 0; ks < 32; ++ks) {
      int k0 = ks * 4 + 2 * g;
      v2f a2, b2;
      a2.x = arow[k0];  a2.y = arow[k0 + 1];
      b2.x = brow[k0];  b2.y = brow[k0 + 1];
      acc = wmma4(a2, b2, acc);
    }
    float bias = b_in[col];
#pragma unroll
    for (int i = 0; i < 8; ++i) {
      int lrow = mt * 16 + i + 8 * g;
      float v = acc[i] + bias;
      if (col < FOUT)            sQ[lrow * FOUT + col] = v * scale;
      else if (col < 2 * FOUT)   sK[lrow * FOUT + (col - FOUT)] = v;
      else                       sV[lrow * FOUT + (col - 2 * FOUT)] = v;
    }
  }
  __syncthreads();

  // phase 3a: scores S[h] = q[h] @ k[h]^T   (64 tiles, 8 per wave, K=32)
  // A[m][k] = sQ[m][h*32+k],  B[k][m'] = sK[m'][h*32+k]
  for (int r = 0; r < 8; ++r) {
    int st = wv + 8 * r;                 // tile id 0..63
    int h = st >> 4, rem = st & 15;
    int mt = rem & 3, ct = rem >> 2;
    const float* __restrict__ arow = sQ + (mt * 16 + ln16) * FOUT + h * 32;
    const float* __restrict__ brow = sK + (ct * 16 + ln16) * FOUT + h * 32;
    v8f acc = {};
    for (int ks = 0; ks < 8; ++ks) {     // K = 32
      int k0 = ks * 4 + 2 * g;
      v2f a2, b2;
      a2.x = arow[k0];  a2.y = arow[k0 + 1];
      b2.x = brow[k0];  b2.y = brow[k0 + 1];
      acc = wmma4(a2, b2, acc);
    }
#pragma unroll
    for (int i = 0; i < 8; ++i) {
      int lrow = mt * 16 + i + 8 * g;
      sS[(h * LQ + lrow) * LQ + ct * 16 + ln16] = acc[i];
    }
  }
  __syncthreads();

  // phase 3b: softmax rows.  thread -> (h = tid/64, l = tid%64)
  {
    const int h = tid >> 6, l = tid & 63;
    float* __restrict__ row = sS + (h * LQ + l) * LQ;
    float mx = -3.4e38f;
    for (int m = 0; m < LQ; ++m) mx = fmaxf(mx, row[m]);
    float sum = 0.0f;
    for (int m = 0; m < LQ; ++m) {
      float p = __expf(row[m] - mx);
      row[m] = p;
      sum += p;
    }
    float inv = 1.0f / sum;
    for (int m = 0; m < LQ; ++m) row[m] *= inv;
  }
  __syncthreads();

  // phase 3c: ctx[h] = P[h] @ v[h]   (32 tiles, 4 per wave, K=64)
  // A[m][k] = sS[(h*64+m)*64+k],  B[k][d'] = sV[k][h*32+d']
  // ctx written into sA (sA not read since the phase-2 barrier)
  for (int r = 0; r < 4; ++r) {
    int st = wv + 8 * r;                 // tile id 0..31
    int h = st >> 3, rem = st & 7;
    int mt = rem >> 1, ct = rem & 1;
    const float* __restrict__ arow = sS + (h * LQ + mt * 16 + ln16) * LQ;
    const float* __restrict__ bcol = sV + h * 32 + ct * 16 + ln16;
    v8f acc = {};
    for (int ks = 0; ks < 16; ++ks) {    // K = 64
      int k0 = ks * 4 + 2 * g;
      v2f a2, b2;
      a2.x = arow[k0];           a2.y = arow[k0 + 1];
      b2.x = bcol[k0 * FOUT];    b2.y = bcol[(k0 + 1) * FOUT];
      acc = wmma4(a2, b2, acc);
    }
#pragma unroll
    for (int i = 0; i < 8; ++i) {
      int lrow = mt * 16 + i + 8 * g;
      sA[lrow * FOUT + h * 32 + ct * 16 + ln16] = acc[i];
    }
  }
  __syncthreads();

  // phase 4: out = ctx @ w_out^T + b_out   (32 tiles, 4 per wave, K=128)
  const int col = wv * 16 + ln16;
  const float bo = b_out[col];
  const float* __restrict__ wrow = w_out + (size_t)col * FOUT;
  for (int mt = 0; mt < 4; ++mt) {
    v8f acc = {};
    const float* __restrict__ arow = sA + (mt * 16 + ln16) * FOUT;
    for (int ks = 0; ks < 32; ++ks) {
      int k0 = ks * 4 + 2 * g;
      v2f a2, b2;
      a2.x = arow[k0];  a2.y = arow[k0 + 1];
      b2.x = wrow[k0];  b2.y = wrow[k0 + 1];
      acc = wmma4(a2, b2, acc);
    }
#pragma unroll
    for (int i = 0; i < 8; ++i) {
      int lrow = mt * 16 + i + 8 * g;
      out[((size_t)(lrow * NNODE + n)) * FOUT + col] = acc[i] + bo;
    }
  }
}

// ---------------------------------------------------------------------------
extern "C" void kernel_launch(void* const* d_in, const int* in_sizes, int n_in,
                              void* d_out, int out_size, void* d_ws, size_t ws_size,
                              hipStream_t stream) {
  const float* x    = (const float*)d_in[0];
  const int*   ei   = (const int*)d_in[1];
  const float* wx0  = (const float*)d_in[2];
  const float* wx1  = (const float*)d_in[3];
  const float* bx   = (const float*)d_in[4];
  // d_in[5], d_in[6] (wh0, wh1) are dead: h0 == 0 so conv(h0) reduces to bias
  const float* bhb  = (const float*)d_in[7];
  const float* w_in = (const float*)d_in[8];
  const float* b_in = (const float*)d_in[9];
  const float* wout = (const float*)d_in[10];
  const float* bout = (const float*)d_in[11];
  float* out = (float*)d_out;

  // workspace layout (floats): deg[1024] | agg[64000*64] | Hn[64000*128]
  float* ws  = (float*)d_ws;
  float* deg = ws;
  float* agg = ws + 1024;
  float* Hn  = ws + 1024 + (size_t)NNODE * TB * FIN;

  const int nzero = 1024 + NNODE * TB * FIN;  // deg + agg
  k_zero<<<(nzero + 255) / 256, 256, 0, stream>>>(ws, nzero);
  k_deg<<<(NEDGE + 255) / 256, 256, 0, stream>>>(ei, deg);
  k_dinv<<<(NNODE + 255) / 256, 256, 0, stream>>>(deg);
  k_scatter<<<NEDGE, 256, 0, stream>>>(ei, x, deg, agg);
  k_gates<<<(NNODE * TB) / 16, 256, 0, stream>>>(x, agg, wx0, wx1, bx, bhb, Hn);
  k_mha<<<NNODE, 256, 0, stream>>>(Hn, w_in, b_in, wout, bout, out);
}